// DescrptSeR_50431505990029
// MI455X (gfx1250) — compile-verified
//
#include <hip/hip_runtime.h>
#include <math.h>

typedef float v2f __attribute__((ext_vector_type(2)));
typedef float v8f __attribute__((ext_vector_type(8)));

#define NNEI_C 138
#define WPB 4           // waves per block (wave32)
#define BLOCK_T 128

// padded LDS geometry
#define K1P 28           // K of stage-2 GEMM (25 -> 28)
#define N1P 64           // N of stage-2 GEMM (50 -> 64)
#define K2P 52           // K of stage-3 GEMM (50 -> 52)
#define N2P 112          // N of stage-3 GEMM (100 -> 112)

__device__ __forceinline__ v8f wmma_f32_16x16x4(v2f a, v2f b, v8f c) {
  // 8 args: (neg_a, A, neg_b, B, c_mod, C, reuse_a, reuse_b)
  return __builtin_amdgcn_wmma_f32_16x16x4_f32(false, a, false, b, (short)0, c,
                                               false, false);
}

__device__ __forceinline__ float smooth_w(float d) {
  const float rs = 0.5f, rc = 6.0f;
  float uu = (d - rs) * (1.0f / (rc - rs));
  float vv = uu * uu * uu * (-6.0f * uu * uu + 15.0f * uu - 10.0f) + 1.0f;
  return (d <= rs) ? 1.0f : ((d >= rc) ? 0.0f : vv);
}

__global__ __launch_bounds__(BLOCK_T) void descrpt_se_r_kernel(
    const float* __restrict__ coord_ext, const int* __restrict__ atype_ext,
    const int* __restrict__ nlist, const float* __restrict__ mean,
    const float* __restrict__ stddev, const float* __restrict__ w0,
    const float* __restrict__ b0, const float* __restrict__ w1,
    const float* __restrict__ b1, const float* __restrict__ w2,
    const float* __restrict__ b2, float* __restrict__ out_res,
    float* __restrict__ out_sw, int NA, int nloc, int nall) {
  // Weights for one neighbor-type at a time, TRANSPOSED + zero-padded so that
  // every WMMA B fragment is one unconditional aligned ds_load_b64.
  __shared__ float W0L[K1P], B0L[K1P];                      // 1x25 -> 28
  __shared__ __align__(16) float W1T[N1P * K1P];            // [n][k] 64x28
  __shared__ float B1L[N1P];
  __shared__ __align__(16) float W2T[N2P * K2P];            // [n][k] 112x52
  __shared__ float B2L[N2P];
  __shared__ float envL[WPB][144];                          // 138 env + pad
  __shared__ __align__(16) float x2L[WPB][16 * K2P];        // 16 rows x 52

  const int tid = threadIdx.x;
  const int wid = tid >> 5;
  const int lane = tid & 31;
  const int half = lane >> 4;   // K/M half of the wave
  const int col = lane & 15;    // N (or row) index within half

  const int a = blockIdx.x * WPB + wid;   // flattened (frame, atom)
  const bool active = a < NA;
  const int f = active ? (a / nloc) : 0;
  const int i = active ? (a % nloc) : 0;

  float* envW = envL[wid];
  float* x2W = x2L[wid];

  // ---------------- Phase A: env + sw ----------------
  if (active) {
    const int at = atype_ext[f * nall + i];
    const float cx = coord_ext[(size_t)(f * nall + i) * 3 + 0];
    const float cy = coord_ext[(size_t)(f * nall + i) * 3 + 1];
    const float cz = coord_ext[(size_t)(f * nall + i) * 3 + 2];
    for (int jj = 0; jj < 5; ++jj) {
      int j = jj * 32 + lane;
      if (j < NNEI_C) {
        int nl = nlist[(size_t)a * NNEI_C + j];
        bool msk = nl >= 0;
        int idx = msk ? nl : 0;
        float dx = coord_ext[(size_t)(f * nall + idx) * 3 + 0] - cx;
        float dy = coord_ext[(size_t)(f * nall + idx) * 3 + 1] - cy;
        float dz = coord_ext[(size_t)(f * nall + idx) * 3 + 2] - cz;
        float lsq = dx * dx + dy * dy + dz * dz;
        float len = sqrtf(msk ? lsq : 1.0f);
        float sw = msk ? smooth_w(len) : 0.0f;
        float env = sw / len;
        env = (env - mean[at * NNEI_C + j]) / stddev[at * NNEI_C + j];
        envW[j] = env;
        out_sw[(size_t)a * NNEI_C + j] = sw;
      }
    }
    if (lane < 6) envW[138 + lane] = 0.0f;
  } else {
    for (int q = lane; q < 144; q += 32) envW[q] = 0.0f;
  }

  float racc[7];
#pragma unroll
  for (int q = 0; q < 7; ++q) racc[q] = 0.0f;

  const int secStart[2] = {0, 46};
  const int secEnd[2] = {46, 138};
  const int ntiles[2] = {3, 6};

  for (int t = 0; t < 2; ++t) {
    __syncthreads();  // previous phase done reading weights
    // Stage weights for this type into LDS (transposed + zero-padded)
    for (int q = tid; q < K1P; q += BLOCK_T) {
      W0L[q] = (q < 25) ? w0[t * 25 + q] : 0.0f;
      B0L[q] = (q < 25) ? b0[t * 25 + q] : 0.0f;
    }
    for (int q = tid; q < N1P * K1P; q += BLOCK_T) {
      int n = q / K1P, k = q % K1P;
      W1T[q] = (n < 50 && k < 25) ? w1[t * 1250 + k * 50 + n] : 0.0f;
    }
    for (int q = tid; q < N1P; q += BLOCK_T)
      B1L[q] = (q < 50) ? b1[t * 50 + q] : 0.0f;
    for (int q = tid; q < N2P * K2P; q += BLOCK_T) {
      int n = q / K2P, k = q % K2P;
      W2T[q] = (n < 100 && k < 50) ? w2[t * 5000 + k * 100 + n] : 0.0f;
    }
    for (int q = tid; q < N2P; q += BLOCK_T)
      B2L[q] = (q < 100) ? b2[t * 100 + q] : 0.0f;
    __syncthreads();

    for (int tile = 0; tile < ntiles[t]; ++tile) {
      const int jbase = secStart[t] + tile * 16;
      const int row = col;  // A-matrix row for this lane
      const float envRow = envW[jbase + row];

      // ---- stage-2 A fragments: x = tanh(env*w0+b0); padded cols give 0
      v2f af[7];
#pragma unroll
      for (int kk = 0; kk < 7; ++kk) {
        int c0 = 4 * kk + 2 * half;
        v2f av;
        av.x = tanhf(envRow * W0L[c0] + B0L[c0]);
        av.y = tanhf(envRow * W0L[c0 + 1] + B0L[c0 + 1]);
        af[kk] = av;
      }

      // ---- stage-2 GEMM: (16x25)@(25x50), N padded to 64
      v8f acc1[4];
#pragma unroll
      for (int nt = 0; nt < 4; ++nt) {
        v8f c = {};
        int n = 16 * nt + col;
        const float* wrow = &W1T[n * K1P];
#pragma unroll
        for (int kk = 0; kk < 7; ++kk) {
          int k0 = 4 * kk + 2 * half;
          v2f bv = *reinterpret_cast<const v2f*>(&wrow[k0]);  // ds_load_b64
          c = wmma_f32_16x16x4(af[kk], bv, c);
        }
        acc1[nt] = c;
      }

      // ---- stage-2 epilogue: x2 = tanh(h1+b1) + concat(x,x) -> LDS (16x52)
#pragma unroll
      for (int nt = 0; nt < 4; ++nt) {
        int n = 16 * nt + col;
        if (n < K2P) {  // cols 50,51 written as 0 pad
          int xc = (n < 25) ? n : (n - 25);  // <= 26, in padded W0L
          float wxc = W0L[xc], bxc = B0L[xc], bn = B1L[n];
#pragma unroll
          for (int v = 0; v < 8; ++v) {
            int m = v + 8 * half;
            float h = tanhf(acc1[nt][v] + bn);
            float xv = tanhf(envW[jbase + m] * wxc + bxc);
            x2W[m * K2P + n] = (n < 50) ? (h + xv) : 0.0f;
          }
        }
      }
      __syncthreads();

      // ---- stage-3 A fragments from x2 (16x50, K padded to 52)
      v2f af2[13];
#pragma unroll
      for (int kk = 0; kk < 13; ++kk) {
        int k0 = 4 * kk + 2 * half;
        af2[kk] = *reinterpret_cast<const v2f*>(&x2W[row * K2P + k0]);
      }

      // ---- stage-3 GEMM: (16x50)@(50x100), N padded to 112
      v8f acc2[7];
#pragma unroll
      for (int nt = 0; nt < 7; ++nt) {
        v8f c = {};
        int n = 16 * nt + col;
        const float* wrow = &W2T[n * K2P];
#pragma unroll
        for (int kk = 0; kk < 13; ++kk) {
          int k0 = 4 * kk + 2 * half;
          v2f bv = *reinterpret_cast<const v2f*>(&wrow[k0]);  // ds_load_b64
          c = wmma_f32_16x16x4(af2[kk], bv, c);
        }
        acc2[nt] = c;
      }

      // ---- stage-3 epilogue: x3 = tanh(h2+b2) + concat(x2,x2); column sums
#pragma unroll
      for (int nt = 0; nt < 7; ++nt) {
        int n = 16 * nt + col;
        float s = 0.0f;
        if (n < 100) {
          int xc = (n < 50) ? n : (n - 50);
          float bn = B2L[n];
#pragma unroll
          for (int v = 0; v < 8; ++v) {
            int m = v + 8 * half;
            bool valid = (jbase + m) < secEnd[t];
            float x3 = tanhf(acc2[nt][v] + bn) + x2W[m * K2P + xc];
            s += valid ? x3 : 0.0f;
          }
        }
        s += __shfl_xor(s, 16, 32);  // combine the two row-halves
        if (half == 0) racc[nt] += s;
      }
      __syncthreads();  // x2W reused by next tile
    }
  }

  // ---------------- Final store: result = sum/NNEI * 0.2 ----------------
  if (active && half == 0) {
    const float scale = 0.2f / (float)NNEI_C;
#pragma unroll
    for (int nt = 0; nt < 7; ++nt) {
      int n = 16 * nt + col;
      if (n < 100) out_res[(size_t)a * 100 + n] = racc[nt] * scale;
    }
  }
}

extern "C" void kernel_launch(void* const* d_in, const int* in_sizes, int n_in,
                              void* d_out, int out_size, void* d_ws,
                              size_t ws_size, hipStream_t stream) {
  const float* coord_ext = (const float*)d_in[0];
  const int* atype_ext = (const int*)d_in[1];
  const int* nlist = (const int*)d_in[2];
  const float* mean = (const float*)d_in[3];
  const float* stddev = (const float*)d_in[4];
  const float* w0 = (const float*)d_in[5];
  const float* b0 = (const float*)d_in[6];
  const float* w1 = (const float*)d_in[7];
  const float* b1 = (const float*)d_in[8];
  const float* w2 = (const float*)d_in[9];
  const float* b2 = (const float*)d_in[10];

  // out = result(nf*nloc*100) ++ sw(nf*nloc*138)
  const int NA = out_size / 238;  // nf * nloc
  const int nf = 2;
  const int nloc = NA / nf;
  const int nall = in_sizes[1] / nf;  // atype_ext is (nf, nall)

  float* out_res = (float*)d_out;
  float* out_sw = (float*)d_out + (size_t)NA * 100;

  const int blocks = (NA + WPB - 1) / WPB;
  descrpt_se_r_kernel<<<blocks, BLOCK_T, 0, stream>>>(
      coord_ext, atype_ext, nlist, mean, stddev, w0, b0, w1, b1, w2, b2,
      out_res, out_sw, NA, nloc, nall);
}